// SwitchableStar_58145267253771
// MI455X (gfx1250) — compile-verified
//
#include <hip/hip_runtime.h>
#include <math.h>

// Problem constants (from reference)
#define NN      2049          // 1 + 16 * 128
#define NP      2112          // padded to 132*16: divisible by 64 for 4x4 tile blocking
#define SUB     128           // 2 * LINE_LEN
#define NBR     16            // N_BRANCH
#define NORD    40            // N_ORDER
#define INV_LC  1.0e9f        // 1 / LC_VAL
#define SIGMA   2.0e-9f       // Krylov scaling (= T3)
#define EXPM_S  256.0f        // 2^8 scaling for expm squaring
#define HORNER_M 10           // Taylor order for exp(X/2^8)
#define TILES   132           // NP / 16
#define RB      4             // 4x4 register blocking: one wave owns 64x64 of C

typedef float v2f __attribute__((ext_vector_type(2)));
typedef float v8f __attribute__((ext_vector_type(8)));

// ---------------------------------------------------------------------------
// prep: a = lds_a * mismatch (switch-gated on row 0), write padded
//   X    = a * (t / 256)     (expm operand, scaled for squaring)
//   Aneg = a * (-sigma)      (Krylov operand)
// ---------------------------------------------------------------------------
__global__ void prep_kernel(const float* __restrict__ lds_a,
                            const float* __restrict__ mism,
                            const int*   __restrict__ sw,
                            const float* __restrict__ tptr,
                            float* __restrict__ X,
                            float* __restrict__ Aneg) {
    int idx = blockIdx.x * blockDim.x + threadIdx.x;
    if (idx >= NP * NP) return;
    int i = idx / NP, j = idx % NP;
    float v = 0.f;
    if (i < NN && j < NN) {
        v = lds_a[i * NN + j] * mism[i * NN + j];
        if (i == 0 && j >= 1 && ((j - 1) % SUB) == 0) {
            int br = (j - 1) / SUB;
            if (br < NBR) v *= (float)sw[br];
        }
    }
    float t = *tptr;
    X[idx]    = v * (t * (1.0f / EXPM_S));
    Aneg[idx] = v * (-SIGMA);
}

// init Krylov vector u0 = b and inp = g0 * u0
__global__ void initvec_kernel(float* __restrict__ u0, float* __restrict__ inp,
                               float g0) {
    int i = blockIdx.x * blockDim.x + threadIdx.x;
    if (i >= NP) return;
    float b = (i == 0) ? INV_LC : 0.f;
    u0[i]  = b;
    inp[i] = g0 * b;
}

// ---------------------------------------------------------------------------
// Krylov step: u_k = (Aneg @ u_{k-1}) / k ;  inp += g_k * u_k
// one workgroup per row, coalesced row read, LDS tree reduction
// ---------------------------------------------------------------------------
__global__ void matvec_kernel(const float* __restrict__ A,
                              const float* __restrict__ x,
                              float* __restrict__ y,
                              float* __restrict__ acc,
                              float inv_k, float gk) {
    __shared__ float red[256];
    const int row = blockIdx.x;
    const float* ar = A + (size_t)row * NP;
    float s = 0.f;
    for (int j = threadIdx.x; j < NP; j += 256) s += ar[j] * x[j];
    red[threadIdx.x] = s;
    __syncthreads();
    for (int off = 128; off > 0; off >>= 1) {
        if (threadIdx.x < off) red[threadIdx.x] += red[threadIdx.x + off];
        __syncthreads();
    }
    if (threadIdx.x == 0) {
        float u = red[0] * inv_k;
        y[row] = u;
        acc[row] += gk * u;
    }
}

// Horner seed: P = X / m + I
__global__ void axpyI_kernel(const float* __restrict__ X, float* __restrict__ P,
                             float scale) {
    int idx = blockIdx.x * blockDim.x + threadIdx.x;
    if (idx >= NP * NP) return;
    int i = idx / NP, j = idx % NP;
    P[idx] = X[idx] * scale + ((i == j) ? 1.f : 0.f);
}

// ---------------------------------------------------------------------------
// WMMA f32 matmul with 4x4 register blocking: C = alpha * (A @ B) + beta * I
// One wave (32 lanes) owns a 64x64 block of C (16 accumulator tiles = 128
// VGPRs). __launch_bounds__(32,1): single wave per EU unlocks the full VGPR
// window (no accumulator spills); latency hiding comes from 16 independent
// WMMAs per K-step. A/B/C are addressed through ONE bumped base pointer each;
// all fragment offsets (r*16*NP*4 <= 405KB, c*64+8448B) fit the signed 24-bit
// VGLOBAL immediate-offset field, so the loop carries only 3 address pairs.
// Operand layouts per CDNA5 ISA 7.12.2 (32-bit A 16x4 / B 4x16 / C 16x16).
// ---------------------------------------------------------------------------
__global__ __launch_bounds__(32, 1)
void wmma_mm_kernel(const float* __restrict__ A,
                    const float* __restrict__ B,
                    float* __restrict__ C,
                    float alpha, float beta) {
    const int tm0  = blockIdx.x * RB;       // base tile row
    const int tn0  = blockIdx.y * RB;       // base tile col
    const int lane = threadIdx.x & 31;
    const int half = lane >> 4;
    const int l15  = lane & 15;

    // Single base pointers (bumped each K-step); fragments via imm offsets.
    const float* ap = A + (size_t)(tm0 * 16 + l15) * NP + 2 * half;
    const float* bp = B + (size_t)(2 * half) * NP + tn0 * 16 + l15;

    v8f acc[RB][RB];
#pragma unroll
    for (int r = 0; r < RB; ++r)
#pragma unroll
        for (int c = 0; c < RB; ++c)
#pragma unroll
            for (int e = 0; e < 8; ++e) acc[r][c][e] = 0.f;

    for (int k = 0; k < NP; k += 4) {
        v2f af[RB];
        v2f bf[RB];
#pragma unroll
        for (int r = 0; r < RB; ++r)                 // A[row+16r][k+2h .. +1]
            af[r] = *(const v2f*)(ap + (size_t)(r * 16) * NP);
#pragma unroll
        for (int c = 0; c < RB; ++c) {
            bf[c].x = bp[c * 16];                    // B[k+2h  ][col+16c]
            bf[c].y = bp[c * 16 + NP];               // B[k+2h+1][col+16c]
        }
#pragma unroll
        for (int r = 0; r < RB; ++r)
#pragma unroll
            for (int c = 0; c < RB; ++c)
                acc[r][c] = __builtin_amdgcn_wmma_f32_16x16x4_f32(
                    /*neg_a=*/false, af[r], /*neg_b=*/false, bf[c],
                    /*c_mod=*/(short)0, acc[r][c],
                    /*reuse_a=*/false, /*reuse_b=*/false);
        ap += 4;
        bp += (size_t)4 * NP;
    }

    // Epilogue: C tile stores through one base pointer + imm offsets.
    float* cp = C + (size_t)(tm0 * 16 + 8 * half) * NP + tn0 * 16 + l15;
#pragma unroll
    for (int r = 0; r < RB; ++r) {
#pragma unroll
        for (int c = 0; c < RB; ++c) {
            const int col = (tn0 + c) * 16 + l15;
#pragma unroll
            for (int e = 0; e < 8; ++e) {
                const int row = (tm0 + r) * 16 + e + 8 * half;
                float v = alpha * acc[r][c][e] + ((row == col) ? beta : 0.f);
                cp[(size_t)(r * 16 + e) * NP + c * 16] = v;
            }
        }
    }
}

// sol = AMP * dot(P[0,:], inp)
__global__ void finaldot_kernel(const float* __restrict__ P,
                                const float* __restrict__ inp,
                                float* __restrict__ out) {
    __shared__ float red[256];
    float s = 0.f;
    for (int j = threadIdx.x; j < NP; j += 256) s += P[j] * inp[j];
    red[threadIdx.x] = s;
    __syncthreads();
    for (int off = 128; off > 0; off >>= 1) {
        if (threadIdx.x < off) red[threadIdx.x] += red[threadIdx.x + off];
        __syncthreads();
    }
    if (threadIdx.x == 0) out[0] = red[0];   // AMP = 1.0
}

// ---------------------------------------------------------------------------
extern "C" void kernel_launch(void* const* d_in, const int* in_sizes, int n_in,
                              void* d_out, int out_size, void* d_ws, size_t ws_size,
                              hipStream_t stream) {
    const float* lds_a = (const float*)d_in[0];
    const float* mism  = (const float*)d_in[1];
    const int*   sw    = (const int*)d_in[2];
    const float* tptr  = (const float*)d_in[3];
    float*       out   = (float*)d_out;

    const size_t MAT = (size_t)NP * NP;
    float* ws   = (float*)d_ws;
    float* X    = ws;                 // expm operand a*t/256
    float* P    = ws + MAT;           // Aneg during Krylov phase, then P ping
    float* Q    = ws + 2 * MAT;       // P pong
    float* u0   = ws + 3 * MAT;
    float* u1   = u0 + NP;
    float* inp  = u1 + NP;

    // Host-side (double precision) combined Taylor coefficients:
    //   inp = sum_k c_k (-a)^k b,  with u_k = (-sigma a)^k b / k!
    //   => g_k = c_k * k! / sigma^k   (all from compile-time constants)
    const double dT1 = 5e-10, dT2 = 1.5e-9, dT3 = 2e-9, SIG = 2e-9;
    double fact[44];
    fact[0] = 1.0;
    for (int i = 1; i < 44; ++i) fact[i] = fact[i - 1] * (double)i;
    float g[NORD + 1];
    for (int k = 0; k <= NORD; ++k) {
        double p1 = pow(dT1, (double)(k + 1));
        double p2 = pow(dT2, (double)(k + 1));
        double p3 = pow(dT3, (double)(k + 1));
        double f1 = p1 * (double)(k + 1) / fact[k + 2];                       // (1/T1)*I1_1 term
        double f2 = (p2 - p1) / fact[k + 1];                                  // I0_2 - I0_1 term
        double f3 = ((p3 * dT3 - p2 * dT2) * (double)(k + 1) / fact[k + 2]
                     - dT3 * (p3 - p2) / fact[k + 1]) / (dT2 - dT3);          // 1/(T2-T3) * (...)
        double c  = f1 + f2 + f3;
        g[k] = (float)(c * fact[k] / pow(SIG, (double)k));
    }

    const int ELEM_BLOCKS = (int)((MAT + 255) / 256);

    // 1) build X = a*t/256 and Aneg = -sigma*a (padded, switch-gated)
    prep_kernel<<<ELEM_BLOCKS, 256, 0, stream>>>(lds_a, mism, sw, tptr, X, P);

    // 2) Krylov chain: inp = sum g_k u_k
    initvec_kernel<<<(NP + 255) / 256, 256, 0, stream>>>(u0, inp, g[0]);
    float* up = u0;
    float* uc = u1;
    for (int k = 1; k <= NORD; ++k) {
        matvec_kernel<<<NP, 256, 0, stream>>>(P, up, uc, inp, 1.0f / (float)k, g[k]);
        float* tmp = up; up = uc; uc = tmp;
    }

    // 3) exp(X) by degree-10 Horner Taylor (P <- I + X*P/j), overwrites Aneg
    axpyI_kernel<<<ELEM_BLOCKS, 256, 0, stream>>>(X, P, 1.0f / (float)HORNER_M);
    dim3 mmGrid(TILES / RB, TILES / RB);   // 33 x 33 waves, no edge cases
    float* Pc = P;
    float* Pn = Q;
    for (int j = HORNER_M - 1; j >= 1; --j) {
        wmma_mm_kernel<<<mmGrid, 32, 0, stream>>>(X, Pc, Pn, 1.0f / (float)j, 1.0f);
        float* tmp = Pc; Pc = Pn; Pn = tmp;
    }

    // 4) 8 squarings: P <- P*P  (undoes the t/256 scaling)
    for (int s = 0; s < 8; ++s) {
        wmma_mm_kernel<<<mmGrid, 32, 0, stream>>>(Pc, Pc, Pn, 1.0f, 0.0f);
        float* tmp = Pc; Pc = Pn; Pn = tmp;
    }

    // 5) sol = row0(expm) . inp
    finaldot_kernel<<<1, 256, 0, stream>>>(Pc, inp, out);
}